// YoloLayer_25872882991901
// MI455X (gfx1250) — compile-verified
//
#include <hip/hip_runtime.h>
#include <stdint.h>

#define NCLS   80
#define NA     3
#define DH     64
#define DW     64
#define NB     32
#define CHA    85                  // 5 + NCLS channels per anchor
#define PLANE  (DH * DW)           // 4096
#define TILE   128                 // cells per block
#define NCELL  (NB * NA * PLANE)   // 393216
#define BOXEL  (NCELL * 6)         // 2359296

__device__ __forceinline__ float sigmoid_fast(float x) {
    return 1.0f / (1.0f + __expf(-x));
}

__global__ __launch_bounds__(TILE) void yolo_decode_kernel(
        const float* __restrict__ in,
        const float* __restrict__ thr_p,
        float* __restrict__ out)
{
    // 85 channel rows x 128 cells = 43,520 bytes of LDS
    __shared__ __align__(16) float smem[CHA * TILE];

    const int tid   = threadIdx.x;
    const int blk   = blockIdx.x;
    const int plane = blk >> 5;               // 32 tiles per (b,a) plane
    const int tpos  = (blk & 31) * TILE;      // cell offset within plane
    const int b     = plane / NA;
    const int a     = plane - b * NA;

    // base of this tile: channel k of this cell-tile lives at gbase + k*PLANE
    const float* gbase = in + (size_t)(b * 255 + a * CHA) * PLANE + tpos;

    // ---- Stage all 85 channel rows into LDS with gfx1250 async copies ----
    // Each issue: 128 lanes x B128 (16B) = 4 channel rows (512B each).
    {
        const int sub  = tid & 31;            // 32 lanes cover one 512B row
        const int koff = tid >> 5;            // 0..3: which of 4 rows
        #pragma unroll
        for (int kk = 0; kk < 22; ++kk) {
            const int k = kk * 4 + koff;
            if (k < CHA) {
                const uint32_t voff  = (uint32_t)((k * PLANE + sub * 4) * 4);
                const uint32_t laddr = (uint32_t)(uintptr_t)(&smem[k * TILE + sub * 4]);
                asm volatile("global_load_async_to_lds_b128 %0, %1, %2"
                             :
                             : "v"(laddr), "v"(voff), "s"(gbase)
                             : "memory");
            }
        }
    }
#if __has_builtin(__builtin_amdgcn_s_wait_asynccnt)
    __builtin_amdgcn_s_wait_asynccnt(0);
#else
    asm volatile("s_wait_asynccnt 0x0" ::: "memory");
#endif
    __syncthreads();

    // ---- Decode one cell per thread, all reads from LDS ----
    const float thr = *thr_p;
    const int   cx  = (tpos + tid) & (DW - 1);
    const int   cy  = (tpos + tid) >> 6;

    // anchors[3,4,5] / stride(32) / grid(64)
    const float aw = (a == 0) ? (0.9375f  / DW) : (a == 1) ? (1.9375f  / DW) : (1.84375f / DW);
    const float ah = (a == 0) ? (1.90625f / DH) : (a == 1) ? (1.40625f / DH) : (3.71875f / DH);

    const float xs  = (sigmoid_fast(smem[0 * TILE + tid]) + (float)cx) * (1.0f / DW);
    const float ys  = (sigmoid_fast(smem[1 * TILE + tid]) + (float)cy) * (1.0f / DH);
    const float wsv = __expf(smem[2 * TILE + tid]) * aw;
    const float hsv = __expf(smem[3 * TILE + tid]) * ah;
    const float det = sigmoid_fast(smem[4 * TILE + tid]);

    // softmax over 80 classes: pass 1 = max/argmax (compares only)
    float m  = smem[5 * TILE + tid];
    int   id = 0;
    #pragma unroll 8
    for (int c = 1; c < NCLS; ++c) {
        const float l = smem[(5 + c) * TILE + tid];
        if (l > m) { m = l; id = c; }
    }
    // pass 2 = sum of exp(l - m); max prob = 1/sum
    float s = 0.0f;
    #pragma unroll 8
    for (int c = 0; c < NCLS; ++c) {
        s += __expf(smem[(5 + c) * TILE + tid] - m);
    }
    const float cmax = 1.0f / s;

    const bool keep = det > thr;
    const int  t    = plane * PLANE + tpos + tid;

    float* bx = out + (size_t)t * 6;
    bx[0] = keep ? xs   : 0.0f;
    bx[1] = keep ? ys   : 0.0f;
    bx[2] = keep ? wsv  : 0.0f;
    bx[3] = keep ? hsv  : 0.0f;
    bx[4] = keep ? det  : 0.0f;
    bx[5] = keep ? cmax : 0.0f;
    out[BOXEL + t]         = (float)id;          // cls_ids as f32 values
    out[BOXEL + NCELL + t] = keep ? 1.0f : 0.0f; // keep mask as f32 0/1
}

extern "C" void kernel_launch(void* const* d_in, const int* in_sizes, int n_in,
                              void* d_out, int out_size, void* d_ws, size_t ws_size,
                              hipStream_t stream) {
    const float* in  = (const float*)d_in[0];
    const float* thr = (const float*)d_in[1];
    float*       out = (float*)d_out;
    const int nblocks = NB * NA * (PLANE / TILE);   // 3072
    yolo_decode_kernel<<<dim3(nblocks), dim3(TILE), 0, stream>>>(in, thr, out);
}